// Brain_2456721293406
// MI455X (gfx1250) — compile-verified
//
#include <hip/hip_runtime.h>
#include <math.h>

#define N_NEURON 100000
#define N_EDGE   10000000
#define N_STEPS  20
#define BATCH    8
#define IN_SIZE  1024
#define OUT_SIZE 1024

#define TILE 32                 // edges per staged tile
#define WAVES_PER_BLOCK 8
#define BLOCK_T 256
#define EDGE_BLOCKS 2048

// ---------------------------------------------------------------------------
// act layout: act[n*8 + b]  (neuron-major, batch-contiguous => 32B per neuron)
// total has the same layout. Both live in d_ws and stay L2-resident (6.4 MB).
// ---------------------------------------------------------------------------

__global__ void init_state_kernel(const float* __restrict__ x,
                                  const float* __restrict__ biases,
                                  float* __restrict__ act,
                                  float* __restrict__ total) {
    int i = blockIdx.x * blockDim.x + threadIdx.x;
    if (i >= N_NEURON * BATCH) return;
    int n = i >> 3, b = i & 7;
    act[i]   = (n < IN_SIZE) ? x[b * IN_SIZE + n] : 0.0f;
    total[i] = biases[n];
}

__device__ __forceinline__ void atomic_add_f32(float* p, float v) {
    __hip_atomic_fetch_add(p, v, __ATOMIC_RELAXED, __HIP_MEMORY_SCOPE_AGENT);
}

// Stage one 32-edge tile (from/to/w, 128B each) into LDS asynchronously.
// Lanes 0-7: from_idx, 8-15: to_idx, 16-23: weights, each lane moves 16B.
__device__ __forceinline__ void stage_tile(unsigned lds_buf_byte,
                                           const int* __restrict__ from_idx,
                                           const int* __restrict__ to_idx,
                                           const float* __restrict__ w,
                                           long long ebase, int lane) {
    int grp = lane >> 3;      // 0=from, 1=to, 2=w, 3=idle
    int sub = lane & 7;       // 16B chunk within the 128B array slab
    if (grp < 3) {
        unsigned long long g;
        if (grp == 0)      g = (unsigned long long)(from_idx + ebase);
        else if (grp == 1) g = (unsigned long long)(to_idx + ebase);
        else               g = (unsigned long long)(w + ebase);
        g += (unsigned)(sub * 16);
        unsigned l = lds_buf_byte + (unsigned)(grp * 128 + sub * 16);
        asm volatile("global_load_async_to_lds_b128 %0, %1, off"
                     :: "v"(l), "v"(g) : "memory");
    }
}

__global__ __launch_bounds__(BLOCK_T)
void edge_kernel(const int* __restrict__ from_idx, const int* __restrict__ to_idx,
                 const float* __restrict__ w, const float* __restrict__ act,
                 float* __restrict__ total) {
    // per wave: 2 buffers x 3 arrays x 32 ints = 768B; 8 waves = 6KB
    __shared__ int smem[WAVES_PER_BLOCK * 2 * 3 * TILE];

    const int lane  = threadIdx.x & 31;
    const int wave  = threadIdx.x >> 5;
    const int gwave = blockIdx.x * WAVES_PER_BLOCK + wave;
    const int nwave = gridDim.x * WAVES_PER_BLOCK;
    const int ntiles = N_EDGE / TILE;   // 312500 (exact)

    int* wbuf = &smem[wave * (2 * 3 * TILE)];
    // low 32 bits of the generic pointer == LDS byte offset within the WG alloc
    unsigned wbuf_byte = (unsigned)(unsigned long long)(void*)wbuf;

    const int b  = lane & 7;    // batch slot
    const int e0 = lane >> 3;   // 0..3: edge sub-slot

    int tile = gwave;
    if (tile >= ntiles) return;

    // prime the pipeline: tile -> buffer 0
    stage_tile(wbuf_byte, from_idx, to_idx, w, (long long)tile * TILE, lane);

    int buf = 0;
    for (; tile < ntiles; tile += nwave) {
        int next = tile + nwave;
        if (next < ntiles) {
            stage_tile(wbuf_byte + (unsigned)((buf ^ 1) * 3 * TILE * 4),
                       from_idx, to_idx, w, (long long)next * TILE, lane);
            asm volatile("s_wait_asynccnt 0x1" ::: "memory");  // current tile done
        } else {
            asm volatile("s_wait_asynccnt 0x0" ::: "memory");
        }

        int* tb = wbuf + buf * 3 * TILE;
#pragma unroll
        for (int k = 0; k < 8; ++k) {
            int el = e0 + k * 4;                       // edge 0..31 in tile
            int fr = tb[el];
            int to = tb[TILE + el];
            float wt = __int_as_float(tb[2 * TILE + el]);
            float v = act[fr * 8 + b] * wt;            // 32B-coalesced over 8 lanes
            atomic_add_f32(&total[to * 8 + b], v);     // L2-resident scatter
        }
        buf ^= 1;
    }
}

// act += tanh(total); clamp inputs; re-seed total with biases for next step.
__global__ void update_act_kernel(const float* __restrict__ x,
                                  const float* __restrict__ biases,
                                  float* __restrict__ total,
                                  float* __restrict__ act) {
    int i = blockIdx.x * blockDim.x + threadIdx.x;
    if (i >= N_NEURON * BATCH) return;
    int n = i >> 3, b = i & 7;
    float a = act[i] + tanhf(total[i]);
    if (n < IN_SIZE) a = x[b * IN_SIZE + n];   // clamp inputs each step
    act[i] = a;
    total[i] = biases[n];                      // ready for next edge pass
}

__global__ void write_out_kernel(const float* __restrict__ act, float* __restrict__ out) {
    int i = blockIdx.x * blockDim.x + threadIdx.x;
    if (i >= BATCH * OUT_SIZE) return;
    int b = i / OUT_SIZE, j = i - b * OUT_SIZE;
    out[i] = act[(N_NEURON - OUT_SIZE + j) * 8 + b];
}

extern "C" void kernel_launch(void* const* d_in, const int* in_sizes, int n_in,
                              void* d_out, int out_size, void* d_ws, size_t ws_size,
                              hipStream_t stream) {
    const float* x        = (const float*)d_in[0];  // [8,1024]
    const float* weights  = (const float*)d_in[1];  // [10M]
    const float* biases   = (const float*)d_in[2];  // [100k]
    const int*   from_idx = (const int*)  d_in[3];  // [10M]
    const int*   to_idx   = (const int*)  d_in[4];  // [10M]

    float* act   = (float*)d_ws;                    // [N*8]  3.2 MB
    float* total = act + (size_t)N_NEURON * BATCH;  // [N*8]  3.2 MB

    const int NB = N_NEURON * BATCH;
    dim3 blk(256);
    dim3 grd((NB + 255) / 256);

    init_state_kernel<<<grd, blk, 0, stream>>>(x, biases, act, total);
    for (int s = 0; s < N_STEPS; ++s) {
        edge_kernel<<<EDGE_BLOCKS, BLOCK_T, 0, stream>>>(from_idx, to_idx, weights, act, total);
        update_act_kernel<<<grd, blk, 0, stream>>>(x, biases, total, act);
    }
    write_out_kernel<<<(BATCH * OUT_SIZE + 255) / 256, 256, 0, stream>>>(act, (float*)d_out);
}